// ModelNew_20624432955429
// MI455X (gfx1250) — compile-verified
//
#include <hip/hip_runtime.h>

typedef float v2f __attribute__((ext_vector_type(2)));
typedef float v8f __attribute__((ext_vector_type(8)));
typedef int   v4i __attribute__((ext_vector_type(4)));

#define T_TOK 16384
#define K_DIM 2048
#define N_DIM 8192
#define N_EXP 8

#define BM 128
#define BN 128
#define BK 16
#define SX 20    // xs row stride in floats (pad: conflict-free, 16B-aligned rows)
#define SW 144   // ws row stride in floats (pad: halves land in disjoint bank groups)

#if __has_builtin(__builtin_amdgcn_global_load_async_to_lds_b128)
#define USE_ASYNC 1
#else
#define USE_ASYNC 0
#endif

typedef v4i __attribute__((address_space(1)))* gptr128;   // global src
typedef v4i __attribute__((address_space(3)))* lptr128;   // LDS dst

__device__ __forceinline__ void copy16_to_lds(const float* g, float* l) {
#if USE_ASYNC
    // GLOBAL_LOAD_ASYNC_TO_LDS_B128: per-lane 16B DMA, tracked by ASYNCcnt
    __builtin_amdgcn_global_load_async_to_lds_b128(
        (gptr128)(void*)g, (lptr128)(void*)l, 0, 0);
#else
    *(float4*)l = *(const float4*)g;
#endif
}

__device__ __forceinline__ void wait_async_copies() {
#if USE_ASYNC
#if __has_builtin(__builtin_amdgcn_s_wait_asynccnt)
    __builtin_amdgcn_s_wait_asynccnt(0);
#else
    asm volatile("s_wait_asynccnt 0x0" ::: "memory");
#endif
#endif
}

__global__ __launch_bounds__(256)
void ragged_gemm_wmma_f32(const float* __restrict__ x,
                          const float* __restrict__ w,
                          const float* __restrict__ bias,
                          const int* __restrict__ gl,
                          float* __restrict__ out)
{
    __shared__ float xs[2][BM * SX];
    __shared__ float ws[2][BK * SW];

    const int e     = blockIdx.z;
    const int start = (e == 0) ? 0 : gl[e - 1];
    const int end   = gl[e];
    const int gsize = end - start;
    if ((int)blockIdx.y * BM >= gsize) return;   // uniform exit: EXEC stays full for WMMA

    const int m0 = start + blockIdx.y * BM;
    const int n0 = blockIdx.x * BN;

    const int tid  = threadIdx.x;
    const int lane = tid & 31;
    const int wave = tid >> 5;
    const int wm   = wave & 1;      // 0..1 : 64-row slot
    const int wn   = wave >> 1;     // 0..3 : 32-col slot
    const int half = lane >> 4;     // 0/1  : lane half
    const int lrow = lane & 15;

    const float* __restrict__ wexp = w + (size_t)e * K_DIM * N_DIM;

    // Issue the global->LDS tile feed for chunk k0 into LDS buffer `buf`.
    auto issue_tiles = [&](int buf, int k0) {
#pragma unroll
        for (int i = 0; i < 2; ++i) {            // x tile: 128x16 f32
            int f   = tid + i * 256;             // 0..511
            int row = f >> 2;                    // 0..127
            int c4  = f & 3;                     // 4 float4 per 16-float row
            int gr  = m0 + row;
            if (gr > T_TOK - 1) gr = T_TOK - 1;  // clamp (rows never stored past `end`)
            copy16_to_lds(x + (size_t)gr * K_DIM + k0 + c4 * 4,
                          &xs[buf][row * SX + c4 * 4]);
        }
#pragma unroll
        for (int i = 0; i < 2; ++i) {            // w tile: 16x128 f32
            int f   = tid + i * 256;
            int row = f >> 5;                    // 0..15
            int c4  = f & 31;                    // 32 float4 per 128-float row
            copy16_to_lds(wexp + (size_t)(k0 + row) * N_DIM + n0 + c4 * 4,
                          &ws[buf][row * SW + c4 * 4]);
        }
    };

    v8f acc[4][2];
#pragma unroll
    for (int mi = 0; mi < 4; ++mi)
#pragma unroll
        for (int ni = 0; ni < 2; ++ni)
            acc[mi][ni] = (v8f){0.f, 0.f, 0.f, 0.f, 0.f, 0.f, 0.f, 0.f};

    issue_tiles(0, 0);

    int buf = 0;
    for (int k0 = 0; k0 < K_DIM; k0 += BK) {
        wait_async_copies();     // my writes into `buf` are in LDS
        __syncthreads();         // everyone's writes visible; everyone done reading buf^1
        if (k0 + BK < K_DIM)
            issue_tiles(buf ^ 1, k0 + BK);   // overlap next tile feed with compute

        const float* __restrict__ xb = xs[buf];
        const float* __restrict__ wb = ws[buf];
#pragma unroll
        for (int kk = 0; kk < BK; kk += 4) {
            v2f a[4];
            v2f b[2];
#pragma unroll
            for (int mi = 0; mi < 4; ++mi) {
                // A 16x4 f32 layout: lanes 0-15 hold K0(v0),K1(v1); lanes 16-31 hold K2,K3
                int row = wm * 64 + mi * 16 + lrow;
                a[mi] = *(const v2f*)&xb[row * SX + kk + 2 * half];
            }
#pragma unroll
            for (int ni = 0; ni < 2; ++ni) {
                // B 4x16 f32 layout: v0: lanes0-15 K=kk, lanes16-31 K=kk+1; v1: K=kk+2, kk+3
                int col = wn * 32 + ni * 16 + lrow;
                b[ni].x = wb[(kk + half) * SW + col];
                b[ni].y = wb[(kk + 2 + half) * SW + col];
            }
#pragma unroll
            for (int mi = 0; mi < 4; ++mi)
#pragma unroll
                for (int ni = 0; ni < 2; ++ni)
                    acc[mi][ni] = __builtin_amdgcn_wmma_f32_16x16x4_f32(
                        false, a[mi], false, b[ni], (short)0, acc[mi][ni],
                        false, false);
        }
        buf ^= 1;
    }

    // Epilogue: C/D layout — vgpr r: lanes0-15 M=r, lanes16-31 M=8+r; N=lane%16.
    // Non-temporal stores: output is write-once (536 MB); keep weights/x in L2.
#pragma unroll
    for (int mi = 0; mi < 4; ++mi) {
        const int rowbase = m0 + wm * 64 + mi * 16 + half * 8;
#pragma unroll
        for (int ni = 0; ni < 2; ++ni) {
            const int col = n0 + wn * 32 + ni * 16 + lrow;
            const float bv = bias[(size_t)e * N_DIM + col];
#pragma unroll
            for (int r = 0; r < 8; ++r) {
                const int row = rowbase + r;
                if (row < end)
                    __builtin_nontemporal_store(acc[mi][ni][r] + bv,
                                                &out[(size_t)row * N_DIM + col]);
            }
        }
    }
}

extern "C" void kernel_launch(void* const* d_in, const int* in_sizes, int n_in,
                              void* d_out, int out_size, void* d_ws, size_t ws_size,
                              hipStream_t stream) {
    (void)in_sizes; (void)n_in; (void)out_size; (void)d_ws; (void)ws_size;
    const float* x    = (const float*)d_in[0];
    const float* w    = (const float*)d_in[1];
    const float* bias = (const float*)d_in[2];
    const int*   gl   = (const int*)d_in[3];
    float* out = (float*)d_out;

    dim3 grid(N_DIM / BN, (T_TOK + BM - 1) / BM, N_EXP);
    ragged_gemm_wmma_f32<<<grid, dim3(256), 0, stream>>>(x, w, bias, gl, out);
}